// APPNP_GC_38981123178701
// MI455X (gfx1250) — compile-verified
//
#include <hip/hip_runtime.h>
#include <hip/hip_bf16.h>
#include <math.h>

typedef __attribute__((ext_vector_type(2))) float v2f;
typedef __attribute__((ext_vector_type(8))) float v8f;

#define H_DIM 128
#define OUT_DIM 10
#define NGRAPH 512
#define KSTEPS 10
#define ALPHA 0.1f

// ---------------------------------------------------------------------------
// Fuse the two linear layers: Wc = W1*W2  [128x128], bc = b1*W2 + b2  [128]
// ---------------------------------------------------------------------------
__global__ __launch_bounds__(128) void fuse_weights_kernel(
    const float* __restrict__ W1, const float* __restrict__ b1,
    const float* __restrict__ W2, const float* __restrict__ b2,
    float* __restrict__ Wc, float* __restrict__ bc) {
  int k = blockIdx.x;      // row of Wc
  int j = threadIdx.x;     // col of Wc
  float acc = 0.f;
  #pragma unroll 8
  for (int t = 0; t < H_DIM; ++t) acc += W1[k * H_DIM + t] * W2[t * H_DIM + j];
  Wc[k * H_DIM + j] = acc;
  if (k == 0) {
    float b = b2[j];
    #pragma unroll 8
    for (int t = 0; t < H_DIM; ++t) b += b1[t] * W2[t * H_DIM + j];
    bc[j] = b;
  }
}

// ---------------------------------------------------------------------------
// x0 = features^T * Wc + bc  via V_WMMA_F32_16X16X4_F32 (fp32 matrix pipe).
// features: [128][N] (feature-major).  out: [N][128] row-major.
// Block = 256 threads = 8 waves; wave w computes the 16x16 tile at
// rows [blockIdx.x*16, +16), cols [w*16, +16).  N is a multiple of 16
// (50000 = 3125*16) so EXEC stays all-ones (WMMA requirement).
// ---------------------------------------------------------------------------
__global__ __launch_bounds__(256) void gemm_x0_wmma_kernel(
    const float* __restrict__ feat, const float* __restrict__ Wc,
    const float* __restrict__ bc, float* __restrict__ out, int Nnodes) {
  const int wave = threadIdx.x >> 5;
  const int lane = threadIdx.x & 31;
  const int colBase = wave * 16;
  const int rowBase = blockIdx.x * 16;
  const int m = lane & 15;            // M for A, N for B/C/D
  const int khalf = (lane >> 4) * 2;  // K sub-block select (0 or 2)
  const int mrow = (lane >> 4) * 8;   // C/D row offset for upper lanes

  // C[m][n] = bc[n] across all 8 accumulator VGPRs
  v8f c;
  const float bias = bc[colBase + m];
  #pragma unroll
  for (int v = 0; v < 8; ++v) c[v] = bias;

  #pragma unroll 4
  for (int k0 = 0; k0 < H_DIM; k0 += 4) {
    v2f a, b;
    // A (16x4, MxK): lane holds A[m][k0+khalf+{0,1}] = feat[k][rowBase+m]
    a.x = feat[(size_t)(k0 + khalf) * Nnodes + rowBase + m];
    a.y = feat[(size_t)(k0 + khalf + 1) * Nnodes + rowBase + m];
    // B (4x16, KxN): lane holds B[k0+khalf+{0,1}][colBase+m] = Wc[k][col]
    b.x = Wc[(k0 + khalf) * H_DIM + colBase + m];
    b.y = Wc[(k0 + khalf + 1) * H_DIM + colBase + m];
    c = __builtin_amdgcn_wmma_f32_16x16x4_f32(
        /*neg_a=*/false, a, /*neg_b=*/false, b,
        /*c_mod=*/(short)0, c, /*reuse_a=*/false, /*reuse_b=*/false);
  }

  // D layout: VGPR v -> row (mrow+v), col m
  #pragma unroll
  for (int v = 0; v < 8; ++v)
    out[(size_t)(rowBase + mrow + v) * H_DIM + colBase + m] = c[v];
}

// ---------------------------------------------------------------------------
// Zero a float buffer
// ---------------------------------------------------------------------------
__global__ void zero_kernel(float* __restrict__ p, long n) {
  long i = (long)blockIdx.x * blockDim.x + threadIdx.x;
  if (i < n) p[i] = 0.f;
}

// ---------------------------------------------------------------------------
// Edge scatter: agg[dst] += w * x[src].  One wave32 per edge; each lane
// owns a float4 chunk of the 128-wide row (coalesced 512B gather from L2,
// 4 global_atomic_add_f32 per lane).
// ---------------------------------------------------------------------------
__global__ __launch_bounds__(256) void scatter_kernel(
    const float* __restrict__ x, const float* __restrict__ ew,
    const int* __restrict__ src, const int* __restrict__ dst,
    float* __restrict__ agg, int E) {
  int edge = blockIdx.x * 8 + (threadIdx.x >> 5);
  if (edge >= E) return;
  const int lane = threadIdx.x & 31;
  const int s = src[edge];
  const int d = dst[edge];
  const float w = ew[edge];
  const float4 v = ((const float4*)(x + (size_t)s * H_DIM))[lane];
  float* ad = agg + (size_t)d * H_DIM + lane * 4;
  atomicAdd(ad + 0, w * v.x);
  atomicAdd(ad + 1, w * v.y);
  atomicAdd(ad + 2, w * v.z);
  atomicAdd(ad + 3, w * v.w);
}

// ---------------------------------------------------------------------------
// x = (1-a)*agg + a*h, and re-zero agg for the next propagation round.
// ---------------------------------------------------------------------------
__global__ void update_kernel(float* __restrict__ x, float* __restrict__ agg,
                              const float* __restrict__ h, long n) {
  long i = (long)blockIdx.x * blockDim.x + threadIdx.x;
  if (i < n) {
    x[i] = (1.f - ALPHA) * agg[i] + ALPHA * h[i];
    agg[i] = 0.f;
  }
}

// ---------------------------------------------------------------------------
// Graph pooling: pooled[batch[node]] += x[node]
// ---------------------------------------------------------------------------
__global__ void pool_kernel(const float* __restrict__ x,
                            const int* __restrict__ batch,
                            float* __restrict__ pooled, long n) {
  long i = (long)blockIdx.x * blockDim.x + threadIdx.x;
  if (i < n) {
    int node = (int)(i >> 7);
    int hh = (int)(i & 127);
    atomicAdd(&pooled[(size_t)batch[node] * H_DIM + hh], x[i]);
  }
}

// ---------------------------------------------------------------------------
// Per-graph head: relu(pooled*V0w+V0b)*V1w+V1b -> log_softmax.
// One 128-thread block per graph; everything staged through LDS.
// ---------------------------------------------------------------------------
__global__ __launch_bounds__(128) void head_kernel(
    const float* __restrict__ pooled, const float* __restrict__ V0w,
    const float* __restrict__ V0b, const float* __restrict__ V1w,
    const float* __restrict__ V1b, float* __restrict__ out) {
  __shared__ float p[H_DIM];
  __shared__ float hid[H_DIM];
  __shared__ float logit[OUT_DIM];
  const int g = blockIdx.x, t = threadIdx.x;

  p[t] = pooled[(size_t)g * H_DIM + t];
  __syncthreads();

  float acc = V0b[t];
  #pragma unroll 8
  for (int k = 0; k < H_DIM; ++k) acc += p[k] * V0w[k * H_DIM + t];
  hid[t] = fmaxf(acc, 0.f);
  __syncthreads();

  if (t < OUT_DIM) {
    float a = V1b[t];
    #pragma unroll 8
    for (int k = 0; k < H_DIM; ++k) a += hid[k] * V1w[k * OUT_DIM + t];
    logit[t] = a;
  }
  __syncthreads();

  if (t < OUT_DIM) {
    float mx = logit[0];
    #pragma unroll
    for (int k = 1; k < OUT_DIM; ++k) mx = fmaxf(mx, logit[k]);
    float sum = 0.f;
    #pragma unroll
    for (int k = 0; k < OUT_DIM; ++k) sum += __expf(logit[k] - mx);
    out[(size_t)g * OUT_DIM + t] = logit[t] - mx - __logf(sum);
  }
}

// ---------------------------------------------------------------------------
extern "C" void kernel_launch(void* const* d_in, const int* in_sizes, int n_in,
                              void* d_out, int out_size, void* d_ws, size_t ws_size,
                              hipStream_t stream) {
  const float* feat = (const float*)d_in[0];   // [128, N]
  const float* ew   = (const float*)d_in[1];   // [E]
  const float* W1   = (const float*)d_in[2];
  const float* b1   = (const float*)d_in[3];
  const float* W2   = (const float*)d_in[4];
  const float* b2   = (const float*)d_in[5];
  const float* V0w  = (const float*)d_in[6];
  const float* V0b  = (const float*)d_in[7];
  const float* V1w  = (const float*)d_in[8];
  const float* V1b  = (const float*)d_in[9];
  const int* eidx   = (const int*)d_in[10];    // [2, E]
  const int* batch  = (const int*)d_in[11];    // [N]

  const int N = in_sizes[0] / H_DIM;           // 50000
  const int E = in_sizes[1];                   // 1600000
  const int* src = eidx;
  const int* dst = eidx + E;

  // workspace carve-out
  char* ws = (char*)d_ws;
  size_t off = 0;
  float* Wc = (float*)(ws + off); off += (size_t)H_DIM * H_DIM * 4;
  float* bc = (float*)(ws + off); off += 256 * 4;
  float* h    = (float*)(ws + off); off += (size_t)N * H_DIM * 4;  // x0 / teleport
  float* x    = (float*)(ws + off); off += (size_t)N * H_DIM * 4;
  float* agg  = (float*)(ws + off); off += (size_t)N * H_DIM * 4;
  float* pooled = (float*)(ws + off); off += (size_t)NGRAPH * H_DIM * 4;

  const long nx = (long)N * H_DIM;

  // 1) fold the two linear layers into one
  fuse_weights_kernel<<<H_DIM, H_DIM, 0, stream>>>(W1, b1, W2, b2, Wc, bc);

  // 2) x0 = feat^T * Wc + bc  (fp32 WMMA)
  gemm_x0_wmma_kernel<<<N / 16, 256, 0, stream>>>(feat, Wc, bc, h, N);

  // 3) APPNP propagation, K rounds
  zero_kernel<<<(int)((nx + 255) / 256), 256, 0, stream>>>(agg, nx);
  for (int k = 0; k < KSTEPS; ++k) {
    const float* xin = (k == 0) ? h : x;
    scatter_kernel<<<(E + 7) / 8, 256, 0, stream>>>(xin, ew, src, dst, agg, E);
    update_kernel<<<(int)((nx + 255) / 256), 256, 0, stream>>>(x, agg, h, nx);
  }

  // 4) pooling + head
  zero_kernel<<<(NGRAPH * H_DIM + 255) / 256, 256, 0, stream>>>(pooled,
                                                               NGRAPH * H_DIM);
  pool_kernel<<<(int)((nx + 255) / 256), 256, 0, stream>>>(x, batch, pooled, nx);
  head_kernel<<<NGRAPH, H_DIM, 0, stream>>>(pooled, V0w, V0b, V1w, V1b,
                                            (float*)d_out);
}